// MCSA_55688545960655
// MI455X (gfx1250) — compile-verified
//
#include <hip/hip_runtime.h>
#include <stdint.h>

// ---------------------------------------------------------------------------
// Fused channel spatial attention for MI455X (gfx1250), wave32.
// fp32 tensor path: V_WMMA_F32_16X16X4_F32.  K/V tiles streamed to LDS with
// the Tensor Data Mover (tensor_load_to_lds + s_wait_tensorcnt), double
// buffered so WMMA consumes LDS while the next tile is in flight.
// ---------------------------------------------------------------------------

typedef float v2f __attribute__((ext_vector_type(2)));
typedef float v8f __attribute__((ext_vector_type(8)));
typedef unsigned int v4u __attribute__((ext_vector_type(4)));
typedef int v4i __attribute__((ext_vector_type(4)));
typedef int v8i __attribute__((ext_vector_type(8)));

#define BB 4
#define CC 64
#define HH 256
#define WW 256
#define HW (HH * WW)
#define SPAD 260     // LDS row stride (floats) for S/Q/P buffers
#define KCW 16       // K-chunk width (k elements per TDM tile)
#define KROWPAD 18   // LDS row stride for K chunk (16 + 2 pad dwords via TDM pad)
#define VCH 16       // V-chunk height (rows per TDM tile)

__device__ __forceinline__ v8f wmma_f32(v2f a, v2f b, v8f c) {
  // (neg_a, A, neg_b, B, c_mod, C, reuse_a, reuse_b)
  return __builtin_amdgcn_wmma_f32_16x16x4_f32(
      false, a, false, b, (short)0, c, false, false);
}

// 2D tile load via Tensor Data Mover.  D# built per CDNA5 ISA sec. 8:
//  group0: count=1 | lds_addr | global_addr | type=2
//  group1: data_size=4B, pad cfg, tensor_dim0=row stride, tensor_dim1=tile_h,
//          tile_dim0=tile_w, tile_dim1=tile_h, tensor_dim0_stride=row stride
__device__ __forceinline__ void tdm_load_2d(unsigned lds_off, const void* gptr,
                                            unsigned tile_w, unsigned tile_h,
                                            unsigned stride_elems,
                                            unsigned pad_en, unsigned pad_ivl,
                                            unsigned pad_amt) {
  unsigned long long ga = (unsigned long long)(uintptr_t)gptr;
  v4u g0;
  g0.x = 1u;                                            // count=1, user mode
  g0.y = lds_off;                                       // lds_addr (bytes)
  g0.z = (unsigned)ga;                                  // global_addr[31:0]
  g0.w = (unsigned)((ga >> 32) & 0x1FFFFFFu) | (2u << 30);  // ga[56:32]|type=2
  v8i g1;
  g1[0] = (int)((2u << 16) | (pad_en << 20) | (pad_ivl << 22) | (pad_amt << 25));
  g1[1] = (int)((stride_elems & 0xFFFFu) << 16);        // tensor_dim0 lo16
  g1[2] = (int)((stride_elems >> 16) | ((tile_h & 0xFFFFu) << 16)); // d0 hi|d1 lo
  g1[3] = (int)((tile_w & 0xFFFFu) << 16);              // tensor_dim1 hi | tile_dim0
  g1[4] = (int)(tile_h & 0xFFFFu);                      // tile_dim1 | tile_dim2=0
  g1[5] = (int)stride_elems;                            // tensor_dim0_stride lo32
  g1[6] = 0;                                            // stride0 hi | stride1 lo
  g1[7] = 0;                                            // stride1 hi
  v4i gz = {0, 0, 0, 0};                                // 2D: groups 2/3 unused
#if __clang_major__ >= 23
  v8i gz8 = {0, 0, 0, 0, 0, 0, 0, 0};
  __builtin_amdgcn_tensor_load_to_lds(g0, g1, gz, gz, gz8, 0);
#else
  __builtin_amdgcn_tensor_load_to_lds(g0, g1, gz, gz, 0);
#endif
}

// ---------------------------------------------------------------------------
// Kernel 1: qkv = w_qkv @ x + b_qkv  (GEMM M=192, K=64, N=HW per batch)
// ---------------------------------------------------------------------------
__global__ __launch_bounds__(128) void qkv_gemm_kernel(
    const float* __restrict__ x, const float* __restrict__ w,
    const float* __restrict__ bias, float* __restrict__ qkv_ws) {
  const int lane = threadIdx.x & 31;
  const int wave = threadIdx.x >> 5;
  const int row  = lane & 15;
  const int hi   = lane >> 4;
  const int koff = hi * 2;
  const int n0   = (blockIdx.x * 4 + wave) * 16;
  const int mt   = blockIdx.y;  // 0..11
  const int bb   = blockIdx.z;
  const int obase = mt * 16;
  const float* xb = x + (size_t)bb * CC * HW;

  v8f acc = {};
  for (int k = 0; k < CC; k += 4) {
    v2f a = *(const v2f*)(w + (size_t)(obase + row) * CC + k + koff);
    v2f bf;
    bf.x = xb[(size_t)(k + koff) * HW + n0 + row];
    bf.y = xb[(size_t)(k + koff + 1) * HW + n0 + row];
    acc = wmma_f32(a, bf, acc);
  }
  const int sel = mt >> 2;  // 0=Q, 1=K, 2=V
  float* dst = qkv_ws + (size_t)sel * BB * CC * HW;
  for (int r = 0; r < 8; ++r) {
    const int o = obase + r + 8 * hi;
    const int c = o - sel * 64;
    dst[((size_t)(bb * CC + c)) * HW + n0 + row] = acc[r] + bias[o];
  }
}

// ---------------------------------------------------------------------------
// Kernel 2: fused  S = Q@K^T -> dwconv3x3+bias -> softmax -> P@V + x
// One workgroup (8 waves) per (b, c, 16-row block).
// K and V tiles arrive in LDS via TDM double buffering.
// ---------------------------------------------------------------------------
__global__ __launch_bounds__(256) void attn_fused_kernel(
    const float* __restrict__ x, const float* __restrict__ wdw,
    const float* __restrict__ bdw, const float* __restrict__ Q,
    const float* __restrict__ Km, const float* __restrict__ V,
    float* __restrict__ out) {
  __shared__ __align__(16) float S_lds[18 * SPAD];         // score halo rows
  __shared__ __align__(16) float QP_lds[18 * SPAD];        // Q rows, later P
  __shared__ __align__(16) float KV_lds[2 * HH * KROWPAD]; // K dbl-buf / V dbl-buf

  const int tid  = threadIdx.x;
  const int lane = tid & 31;
  const int wave = tid >> 5;
  const int row  = lane & 15;
  const int hi   = lane >> 4;
  const int koff = hi * 2;
  const int r0   = blockIdx.x * 16;
  const int c    = blockIdx.y;
  const int bb   = blockIdx.z;
  const size_t base = ((size_t)(bb * CC + c)) * HW;
  const float* q = Q + base;
  const float* k = Km + base;
  const float* v = V + base;
  const unsigned kv_off = (unsigned)(uintptr_t)&KV_lds[0];  // LDS byte offset

  // ---- Stage 1: S rows r0-1 .. r0+16  (two overlapping 16-row WMMA tiles)
  // Kick off K chunk 0 via TDM, then stage Q halo rows into LDS (coalesced).
  if (wave == 0) tdm_load_2d(kv_off, k, KCW, HH, WW, 1, 3, 1);
  for (int r = 0; r < 18; ++r) {
    const int gr = r0 - 1 + r;  // zero-pad rows outside the image ('SAME')
    QP_lds[r * SPAD + tid] = (gr >= 0 && gr < HH) ? q[(size_t)gr * WW + tid] : 0.0f;
  }

  v8f acc[4] = {};
  for (int ch = 0; ch < WW / KCW; ++ch) {
    if (wave == 0) {
      if (ch + 1 < WW / KCW) {
        tdm_load_2d(kv_off + ((ch + 1) & 1) * (HH * KROWPAD * 4),
                    k + (ch + 1) * KCW, KCW, HH, WW, 1, 3, 1);
        __builtin_amdgcn_s_wait_tensorcnt(1);  // chunk ch has landed
      } else {
        __builtin_amdgcn_s_wait_tensorcnt(0);
      }
    }
    __syncthreads();
    const float* kb = &KV_lds[(ch & 1) * (HH * KROWPAD)];
    const int kbase = ch * KCW;
    for (int t = 0; t < 4; ++t) {
      const int tile = wave + 8 * t;          // 32 tiles: (rowTile, gTile)
      const int lr   = (tile & 1) * 2 + row;  // local Q/S row (0..17)
      const int g0   = (tile >> 1) * 16;
      for (int kl = 0; kl < KCW; kl += 4) {
        v2f a  = *(const v2f*)&QP_lds[lr * SPAD + kbase + kl + koff];
        v2f bf = *(const v2f*)&kb[(g0 + row) * KROWPAD + kl + koff];
        acc[t] = wmma_f32(a, bf, acc[t]);
      }
    }
    __syncthreads();
  }
  for (int t = 0; t < 4; ++t) {
    const int tile  = wave + 8 * t;
    const int sbase = (tile & 1) * 2;
    const int g0    = (tile >> 1) * 16;
    for (int r = 0; r < 8; ++r)
      S_lds[(sbase + r + 8 * hi) * SPAD + g0 + row] = acc[t][r];  // dup writes identical
  }
  __syncthreads();

  // Prefetch V chunk 0 (overlaps the softmax stage; K buffers are dead now).
  if (wave == 0) tdm_load_2d(kv_off, v, WW, VCH, WW, 0, 0, 0);

  // ---- Stage 2: depthwise 3x3 + bias, row softmax -> P (aliases Q region)
  float w9[9];
#pragma unroll
  for (int i = 0; i < 9; ++i) w9[i] = wdw[c * 9 + i];
  const float bc = bdw[c];

  for (int rl = wave; rl < 16; rl += 8) {
    const int i = rl + 1;
    float vals[8];
    float mx = -3.4e38f;
#pragma unroll
    for (int j = 0; j < 8; ++j) {
      const int g = j * 32 + lane;  // stride-1 across lanes -> conflict-free
      float a = bc;
#pragma unroll
      for (int di = -1; di <= 1; ++di)
#pragma unroll
        for (int dg = -1; dg <= 1; ++dg) {
          const int gg = g + dg;
          const float sv = (gg >= 0 && gg < WW) ? S_lds[(i + di) * SPAD + gg] : 0.0f;
          a += w9[(di + 1) * 3 + (dg + 1)] * sv;
        }
      vals[j] = a;
      mx = fmaxf(mx, a);
    }
#pragma unroll
    for (int m = 16; m >= 1; m >>= 1) mx = fmaxf(mx, __shfl_xor(mx, m, 32));
    float sum = 0.0f;
#pragma unroll
    for (int j = 0; j < 8; ++j) {
      vals[j] = __expf(vals[j] - mx);
      sum += vals[j];
    }
#pragma unroll
    for (int m = 16; m >= 1; m >>= 1) sum += __shfl_xor(sum, m, 32);
    const float inv = 1.0f / sum;
#pragma unroll
    for (int j = 0; j < 8; ++j) QP_lds[rl * SPAD + j * 32 + lane] = vals[j] * inv;
  }
  __syncthreads();

  // ---- Stage 3: out = P @ V + x   (V streamed in 16-row TDM chunks)
  v8f accn[2] = {};
  for (int ch = 0; ch < HH / VCH; ++ch) {
    if (wave == 0) {
      if (ch + 1 < HH / VCH) {
        tdm_load_2d(kv_off + ((ch + 1) & 1) * (VCH * WW * 4),
                    v + (size_t)(ch + 1) * VCH * WW, WW, VCH, WW, 0, 0, 0);
        __builtin_amdgcn_s_wait_tensorcnt(1);
      } else {
        __builtin_amdgcn_s_wait_tensorcnt(0);
      }
    }
    __syncthreads();
    const float* vb = &KV_lds[(ch & 1) * (VCH * WW)];
    for (int t = 0; t < 2; ++t) {
      const int n0 = (wave + 8 * t) * 16;
      for (int kl = 0; kl < VCH; kl += 4) {
        v2f a = *(const v2f*)&QP_lds[row * SPAD + ch * VCH + kl + koff];
        v2f bf;
        bf.x = vb[(kl + koff) * WW + n0 + row];
        bf.y = vb[(kl + koff + 1) * WW + n0 + row];
        accn[t] = wmma_f32(a, bf, accn[t]);
      }
    }
    __syncthreads();
  }
  for (int t = 0; t < 2; ++t) {
    const int n0 = (wave + 8 * t) * 16;
    for (int r = 0; r < 8; ++r) {
      const int i = r + 8 * hi;
      const size_t idx = base + (size_t)(r0 + i) * WW + n0 + row;
      out[idx] = accn[t][r] + x[idx];
    }
  }
}

// ---------------------------------------------------------------------------
extern "C" void kernel_launch(void* const* d_in, const int* in_sizes, int n_in,
                              void* d_out, int out_size, void* d_ws,
                              size_t ws_size, hipStream_t stream) {
  (void)in_sizes; (void)n_in; (void)out_size; (void)ws_size;
  const float* x     = (const float*)d_in[0];
  const float* w_qkv = (const float*)d_in[1];
  const float* b_qkv = (const float*)d_in[2];
  const float* w_dw  = (const float*)d_in[3];
  const float* b_dw  = (const float*)d_in[4];
  float* out = (float*)d_out;
  float* ws  = (float*)d_ws;

  const size_t plane = (size_t)BB * CC * HW;
  float* Q = ws;
  float* K = ws + plane;
  float* V = ws + 2 * plane;

  dim3 g1(HW / 64, 12, BB);
  qkv_gemm_kernel<<<g1, 128, 0, stream>>>(x, w_qkv, b_qkv, ws);

  dim3 g2(HH / 16, CC, BB);
  attn_fused_kernel<<<g2, 256, 0, stream>>>(x, w_dw, b_dw, Q, K, V, out);
}